// DNMPScene_29197187678881
// MI455X (gfx1250) — compile-verified
//
#include <hip/hip_runtime.h>

// CDNA5 / gfx1250, wave32. One wave per ray iteration.
// take_along_axis (depth sort) executed as an exact f32 permutation-matrix
// GEMM on V_WMMA_F32_16X16X4_F32. All lane-static control flow hoisted out
// of the ray loop; stores are non-temporal to preserve L2 residency of the
// ~112 MB gather working set (fits in the 192 MB L2).

typedef __attribute__((ext_vector_type(2))) float v2f;
typedef __attribute__((ext_vector_type(8))) float v8f;

__global__ __launch_bounds__(256)
void dnmp_render_sample_wmma(
    const float* __restrict__ bary,   // [P,4,3]
    const float* __restrict__ zbuf,   // [P,4]
    const float* __restrict__ verts,  // [V,3]
    const float* __restrict__ feats,  // [V,24]
    const float* __restrict__ norms,  // [V,3]
    const int*   __restrict__ p2f,    // [P,4]
    const int*   __restrict__ faces,  // [F,3]
    const int*   __restrict__ pix,    // [R]
    float* __restrict__ out,          // tex | pts | dep | nrm (flat)
    int R)
{
  const int lane = threadIdx.x & 31;
  const int half = lane >> 4;         // 16-lane half
  const int n    = lane & 15;         // B/D column == A row M
  const int gwave = blockIdx.x * (blockDim.x >> 5) + (threadIdx.x >> 5);
  const int nwave = gridDim.x * (blockDim.x >> 5);
  const int sA = half * 2;            // slot in VGPR0 (K = vgpr + 2*half)

  // ---- lane-static: attribute source for dim group 1 (d1 = 16+n) ----
  // packed column d: [0,24) feats, [24,27) verts, [27,30) norms, [30,32) pad
  const int d1 = 16 + n;
  const float* tab1; int st1, col1; float v1ok;
  if (d1 < 24)      { tab1 = feats; st1 = 24; col1 = d1;      v1ok = 1.f; }
  else if (d1 < 27) { tab1 = verts; st1 = 3;  col1 = d1 - 24; v1ok = 1.f; }
  else if (d1 < 30) { tab1 = norms; st1 = 3;  col1 = d1 - 27; v1ok = 1.f; }
  else              { tab1 = feats; st1 = 0;  col1 = 0;       v1ok = 0.f; }

  // ---- lane-static: store target for D1 rows ----
  // tex: off = r*96 + j*24 + d1            (n < 8)
  // pts: off = R*96  + r*12 + j*3 + (n-8)  (8 <= n < 11)
  // nrm: off = R*112 + r*12 + j*3 + (n-11) (11 <= n < 14)
  int s1base, s1rs, s1js;
  const bool s1ok = (half == 0) && (n < 14);
  if (n < 8)       { s1base = d1;              s1rs = 96; s1js = 24; }
  else if (n < 11) { s1base = R*96  + (n-8);   s1rs = 12; s1js = 3;  }
  else             { s1base = R*112 + (n-11);  s1rs = 12; s1js = 3;  }
  const bool s0ok = (half == 0);
  const int depBase = R * 108;

  for (int r = gwave; r < R; r += nwave) {   // wave-uniform trip count
    const int p = pix[r];

    // ---- per-ray header: one b128 each (rows are 16B aligned) ----
    const int4   rawv = *(const int4*)  (p2f  + (size_t)p * 4);
    const float4 dzv  = *(const float4*)(zbuf + (size_t)p * 4);
    const int   raw[4] = {rawv.x, rawv.y, rawv.z, rawv.w};
    const float dz[4]  = {dzv.x,  dzv.y,  dzv.z,  dzv.w};

    int fidx[4]; float dep[4]; float live[4];
#pragma unroll
    for (int k = 0; k < 4; ++k) {
      const bool inv = raw[k] < 0;            // invalid -> fall back to slot 0
      const int  f   = inv ? raw[0] : raw[k];
      dep[k]  = inv ? dz[0] : dz[k];          // depth falls back, is NOT masked
      const bool m = f < 0;                   // still -1 -> zero attributes
      live[k] = m ? 0.f : 1.f;
      fidx[k] = m ? 0 : f;
    }
    // stable argsort ranks (matches jnp.argsort tie-breaking)
    int rank[4];
#pragma unroll
    for (int k = 0; k < 4; ++k) {
      int rk = 0;
#pragma unroll
      for (int j = 0; j < 4; ++j)
        rk += (dep[j] < dep[k]) || (dep[j] == dep[k] && j < k);
      rank[k] = rk;
    }

    // ---- this lane's two slots: fallback bary + face vertex ids ----
    float bw[2][3]; int vid[2][3];
#pragma unroll
    for (int t = 0; t < 2; ++t) {
      const int s  = sA + t;
      const int bs = (raw[s] < 0) ? 0 : s;
#pragma unroll
      for (int v = 0; v < 3; ++v)
        bw[t][v] = bary[((size_t)p * 4 + bs) * 3 + v];
      const int f = fidx[s];
#pragma unroll
      for (int v = 0; v < 3; ++v)
        vid[t][v] = faces[(size_t)f * 3 + v];
    }

    // ---- B values: branchless gathers + select (invalid faces are rare) ----
    // group 0: d = n in [0,16)  -> always feats column n (coalesced 64B/half)
    // group 1: d = 16+n         -> lane-static table tab1/st1/col1
    float b0[2], b1[2];
#pragma unroll
    for (int t = 0; t < 2; ++t) {
      const float g0 = bw[t][0] * feats[(size_t)vid[t][0] * 24 + n]
                     + bw[t][1] * feats[(size_t)vid[t][1] * 24 + n]
                     + bw[t][2] * feats[(size_t)vid[t][2] * 24 + n];
      const float g1 = bw[t][0] * tab1[(size_t)vid[t][0] * st1 + col1]
                     + bw[t][1] * tab1[(size_t)vid[t][1] * st1 + col1]
                     + bw[t][2] * tab1[(size_t)vid[t][2] * st1 + col1];
      b0[t] = live[sA + t] * g0;
      b1[t] = live[sA + t] * v1ok * g1;
    }

    // A (16x4): one-hot permutation, A[m,k] = (rank[k]==m). Exact in f32.
    v2f A;  A.x  = (rank[sA]     == n) ? 1.f : 0.f;
            A.y  = (rank[sA + 1] == n) ? 1.f : 0.f;
    v2f B0; B0.x = b0[0]; B0.y = b0[1];
    v2f B1; B1.x = b1[0]; B1.y = b1[1];
    v8f Cz = {0.f, 0.f, 0.f, 0.f, 0.f, 0.f, 0.f, 0.f};

    // D[m,n] = sum_k [rank[k]==m] * interp(k, dim) == depth-sorted attrs
    v8f D0 = __builtin_amdgcn_wmma_f32_16x16x4_f32(
        false, A, false, B0, (short)0, Cz, false, false);
    v8f D1 = __builtin_amdgcn_wmma_f32_16x16x4_f32(
        false, A, false, B1, (short)0, Cz, false, false);

    // ---- stores (non-temporal: written once, keep L2 for the gather set) ----
    if (s0ok) {                         // lanes 0-15: D row j = sorted slot j
      const int off0 = r * 96 + n;      // tex dims 0..15, 64B coalesced
#pragma unroll
      for (int j = 0; j < 4; ++j)
        __builtin_nontemporal_store(D0[j], out + off0 + j * 24);
    }
    if (s1ok) {                         // dims 16..29 -> tex/pts/nrm
      const int off1 = s1base + r * s1rs;
#pragma unroll
      for (int j = 0; j < 4; ++j)
        __builtin_nontemporal_store(D1[j], out + off1 + j * s1js);
    }
    if (n < 2) {                        // lanes 0,1 -> slots 0,1; 16,17 -> 2,3
      const int s = sA + n;
      __builtin_nontemporal_store(dep[s], out + depBase + r * 4 + rank[s]);
    }
  }
}

extern "C" void kernel_launch(void* const* d_in, const int* in_sizes, int n_in,
                              void* d_out, int out_size, void* d_ws, size_t ws_size,
                              hipStream_t stream) {
  const float* bary  = (const float*)d_in[0];
  const float* zbuf  = (const float*)d_in[1];
  const float* verts = (const float*)d_in[2];
  const float* feats = (const float*)d_in[3];
  const float* norms = (const float*)d_in[4];
  const int*   p2f   = (const int*)d_in[5];
  const int*   faces = (const int*)d_in[6];
  const int*   pix   = (const int*)d_in[7];
  const int R = in_sizes[7];

  dnmp_render_sample_wmma<<<4096, 256, 0, stream>>>(
      bary, zbuf, verts, feats, norms, p2f, faces, pix, (float*)d_out, R);
}